// NgramAttentions_45380624450151
// MI455X (gfx1250) — compile-verified
//
#include <hip/hip_runtime.h>
#include <hip/hip_bf16.h>
#include <math.h>

// NgramAttentions fused single-pass kernel for MI455X (gfx1250, wave32).
//
// C=8, L=4096, N=4096, D=128, V=100000.
// out[l, c*128+d] = sum_n attn[c,l,n]*emb[c,n,d],
//   attn = exp(u)*clip(mask) / (rowsum + 1e-10), u = (hidden @ emb^T)/sqrt(D)
// Channel softmax over a size-1 axis == 1.0 -> omitted.
//
// Flash-style fusion: never materialize u/attn (would be 512MB each in HBM).
// Both GEMMs use V_WMMA_F32_16X16X4_F32 (fp32 in == fp32 math, memory-bound
// problem: dominant traffic is the 512MB mask read).

typedef float v2f __attribute__((ext_vector_type(2)));
typedef float v8f __attribute__((ext_vector_type(8)));

#define C_CH   8
#define L_DIM  4096
#define N_DIM  4096
#define D_DIM  128
#define NT     16                 // N tile per iteration
#define EMB_STRIDE (D_DIM + 4)    // padded LDS row stride (floats), gcd(132,64banks) small
#define P_STRIDE   17             // padded p-tile stride

__global__ __launch_bounds__(256)
void ngram_attn_fused(const int*   __restrict__ ngram_seq,   // [C, N]
                      const float* __restrict__ hidden,      // [L, D]
                      const float* __restrict__ mask,        // [C, L, N]
                      const float* __restrict__ W_emb,       // [V, D] (row 0 already zero)
                      float*       __restrict__ out)         // [L, C*D]
{
    __shared__ float emb_lds[NT * EMB_STRIDE];        // 16 x 132 floats (8.4 KB)
    __shared__ float p_lds[8][16 * P_STRIDE];         // per-wave 16x17 (8.7 KB)

    const int tid  = threadIdx.x;
    const int lane = tid & 31;
    const int wave = tid >> 5;

    const int c  = blockIdx.x & 7;                    // channel
    const int lb = blockIdx.x >> 3;                   // 128-row L slab
    const int l0 = lb * 128 + wave * 16;              // this wave's 16 L rows

    const int half = lane >> 4;                       // 0 or 1 (lane half)
    const int l16  = lane & 15;

    // ---- resident hidden A-fragments: 16 L rows x K=128 ----
    // f32 16x4 A layout: lanes 0-15 hold M=lane; VGPR0 = K0 | K2(halves),
    // VGPR1 = K1 | K3  => per lane: k = 4*kc + 2*half, contiguous float2.
    v2f afrag[32];
    {
        const float* hrow = hidden + (size_t)(l0 + l16) * D_DIM;
        #pragma unroll
        for (int kc = 0; kc < 32; ++kc) {
            const int k = kc * 4 + half * 2;
            afrag[kc].x = hrow[k];
            afrag[kc].y = hrow[k + 1];
        }
    }

    v8f acc[8];                                       // 16L x 128D accumulator
    #pragma unroll
    for (int d = 0; d < 8; ++d) acc[d] = {};
    float rs[8];                                      // per-lane partial row sums
    #pragma unroll
    for (int r = 0; r < 8; ++r) rs[r] = 0.0f;

    const float inv_temper = 0.08838834764831845f;    // 1/sqrt(128)
    const int*   seq   = ngram_seq + c * N_DIM;
    const float* mbase = mask + (size_t)c * L_DIM * N_DIM;

    for (int n0 = 0; n0 < N_DIM; n0 += NT) {
        __syncthreads();                              // previous tile consumed
        // ---- cooperative emb gather: 16 rows x 128 cols, 8 floats/thread ----
        {
            const int n   = tid >> 4;                 // 0..15
            const int col = (tid & 15) * 8;           // 0..120
            const int idx = seq[n0 + n];
            const float4* src = (const float4*)(W_emb + (size_t)idx * D_DIM + col);
            float4 v0 = src[0];
            float4 v1 = src[1];
            float4* dst = (float4*)(&emb_lds[n * EMB_STRIDE + col]);
            dst[0] = v0;
            dst[1] = v1;
        }
        __syncthreads();                              // tile ready

        // prefetch next mask tile (emits global_prefetch_b8); uniform branch.
        if (n0 + NT < N_DIM) {
            __builtin_prefetch(mbase + (size_t)(l0 + 8 * half) * N_DIM + n0 + NT + l16, 0, 0);
        }

        // ---- u(16L x 16N) = hidden @ emb^T, K = 128 ----
        v8f u = {};
        #pragma unroll
        for (int kc = 0; kc < 32; ++kc) {
            const int k = kc * 4 + half * 2;
            v2f b;                                    // B = emb^T: B[k][n] = emb[n][k]
            b.x = emb_lds[l16 * EMB_STRIDE + k];
            b.y = emb_lds[l16 * EMB_STRIDE + k + 1];
            u = __builtin_amdgcn_wmma_f32_16x16x4_f32(false, afrag[kc], false, b,
                                                      (short)0, u, false, false);
        }

        // ---- p = exp(u/temper) * clip(mask,0,1); accumulate row sums ----
        // C/D layout: row M = r + 8*half, col Nn = l16.
        float p_sc[8];
        #pragma unroll
        for (int r = 0; r < 8; ++r) {
            const int lrow = l0 + r + 8 * half;
            float mv = mbase[(size_t)lrow * N_DIM + n0 + l16];
            mv = fminf(fmaxf(mv, 0.0f), 1.0f);
            float pv = __expf(u[r] * inv_temper) * mv;
            p_sc[r] = pv;
            rs[r]  += pv;
        }

        // ---- relayout p via per-wave LDS (C-layout -> A-layout) ----
        float* pw = p_lds[wave];
        #pragma unroll
        for (int r = 0; r < 8; ++r)
            pw[(r + 8 * half) * P_STRIDE + l16] = p_sc[r];
        // same-wave LDS: DScnt keeps stores before dependent loads in order

        // ---- acc(16L x 128D) += p(16x16) @ emb(16N x 128D), K = 16 ----
        #pragma unroll
        for (int kc = 0; kc < 4; ++kc) {
            const int k = kc * 4 + half * 2;
            v2f a;                                    // A = p, row m = l16
            a.x = pw[l16 * P_STRIDE + k];
            a.y = pw[l16 * P_STRIDE + k + 1];
            #pragma unroll
            for (int d = 0; d < 8; ++d) {
                v2f b;                                // B[k][dcol] = emb[k][d*16+l16]
                b.x = emb_lds[k * EMB_STRIDE + d * 16 + l16];
                b.y = emb_lds[(k + 1) * EMB_STRIDE + d * 16 + l16];
                acc[d] = __builtin_amdgcn_wmma_f32_16x16x4_f32(false, a, false, b,
                                                               (short)0, acc[d], false, false);
            }
        }
    }

    // ---- reduce row sums across the 16 lanes of each half-wave ----
    #pragma unroll
    for (int r = 0; r < 8; ++r) {
        float s = rs[r];
        s += __shfl_xor(s, 1, 32);
        s += __shfl_xor(s, 2, 32);
        s += __shfl_xor(s, 4, 32);
        s += __shfl_xor(s, 8, 32);
        rs[r] = 1.0f / (s + 1e-10f);
    }

    // channel softmax over size-1 axis == 1.0 -> identity scaling.
    // ---- out[l, c*128 + d], coalesced over l16 ----
    #pragma unroll
    for (int r = 0; r < 8; ++r) {
        const int lrow = l0 + r + 8 * half;
        float* orow = out + (size_t)lrow * (C_CH * D_DIM) + c * D_DIM;
        #pragma unroll
        for (int d = 0; d < 8; ++d)
            orow[d * 16 + l16] = acc[d][r] * rs[r];
    }
}

extern "C" void kernel_launch(void* const* d_in, const int* in_sizes, int n_in,
                              void* d_out, int out_size, void* d_ws, size_t ws_size,
                              hipStream_t stream) {
    // setup_inputs order:
    // 0: ngram_seq [C,N] int, 1: hidden_state [L,D] f32, 2: mask [C,L,N] f32,
    // 3: channel_ids [C] int (unused), 4: W_emb [V,D] f32, 5: W_channel [C,1] f32
    //    (unused: softmax over size-1 axis is identically 1)
    const int*   ngram_seq = (const int*)  d_in[0];
    const float* hidden    = (const float*)d_in[1];
    const float* mask      = (const float*)d_in[2];
    const float* W_emb     = (const float*)d_in[4];
    float*       out       = (float*)d_out;

    dim3 grid(C_CH * (L_DIM / 128));   // 256 blocks: (channel, 128-row L slab)
    dim3 block(256);                   // 8 waves x 16 L rows
    ngram_attn_fused<<<grid, block, 0, stream>>>(ngram_seq, hidden, mask, W_emb, out);
}